// MDNDNNLoss_65704409694903
// MI455X (gfx1250) — compile-verified
//
#include <hip/hip_runtime.h>
#include <cstdint>

#define B_ 64
#define L_ 256
#define T_ 1600
#define TC 16                 // t-steps per LDS chunk
#define NCHUNK (T_ / TC)      // 100
#define LPST 20               // padded LDS row stride in floats (80B: 16B-aligned rows, few bank conflicts)
#define NEGV (-1e30f)

// Raw-hardware transcendentals (safe here: args never denormal, exp arg <= 0).
__device__ __forceinline__ float flog(float x) {   // natural log via v_log_f32
  return __builtin_amdgcn_logf(x) * 0.69314718055994530942f;
}
__device__ __forceinline__ float fexp(float x) {   // natural exp via v_exp_f32
  return __builtin_amdgcn_exp2f(x * 1.44269504088896340736f);
}
// logaddexp matching jnp semantics: max + log1p(exp(min-max))
__device__ __forceinline__ float lae(float a, float b) {
  float m = fmaxf(a, b);
  float d = fminf(a, b) - m;          // <= 0, exp never overflows
  return m + flog(1.0f + fexp(d));
}

// ---------------- Kernel 1: log_prob_matrix = log(probs + 1e-30) ----------------
// Bandwidth-bound: ~210MB traffic -> ~9us at 23.3 TB/s.
__global__ void __launch_bounds__(256) mdn_log_kernel(
    const float4* __restrict__ probs4, float* __restrict__ out, int n4) {
  int i = blockIdx.x * 256 + threadIdx.x;
  if (i < n4) {
    float4 v = probs4[i];
    // out is d_out+1 (4B aligned only) -> scalar stores
    out[4 * i + 0] = flog(v.x + 1e-30f);
    out[4 * i + 1] = flog(v.y + 1e-30f);
    out[4 * i + 2] = flog(v.z + 1e-30f);
    out[4 * i + 3] = flog(v.w + 1e-30f);
  }
}

// ---------------- Kernel 2: monotonic-alignment DP ----------------
// One block per batch, thread l owns label l. Two time steps per workgroup
// barrier (K=2 blocking). probs tiles staged via CDNA5 async global->LDS DMA
// (double-buffered), fenced with s_wait_asynccnt.
__global__ void __launch_bounds__(256) mdn_dp_kernel(
    const float* __restrict__ probs,
    const int* __restrict__ text_len,
    const int* __restrict__ mel_len,
    float* __restrict__ alpha_out) {
  __shared__ __align__(16) float lpbuf[2][L_ * LPST];  // 2 x 256 x 20 floats = 40KB
  __shared__ float sbuf[2][L_];                        // DP state double buffer

  const int b = blockIdx.x;
  const int tid = threadIdx.x;                 // == label l
  const float* pb = probs + (size_t)b * L_ * T_;
  const int tl = text_len[b] - 1;
  const int ml = mel_len[b] - 1;

  unsigned ldsbase0 = (unsigned)(uintptr_t)&lpbuf[0][0];
  unsigned ldsbase1 = (unsigned)(uintptr_t)&lpbuf[1][0];

  // stage probs[b, 0..255, t0..t0+TC) into lpbuf[buf] (row-padded), async DMA
  auto issue_fill = [&](int buf, int t0) {
    const float* base = pb + t0;
    unsigned lbase = buf ? ldsbase1 : ldsbase0;
#pragma unroll
    for (int i = 0; i < 4; ++i) {
      int idx = i * 256 + tid;                 // 1024 float4 tiles: 256 rows x 4
      int row = idx >> 2;
      int c4  = idx & 3;
      const float* g = base + row * T_ + c4 * 4;                     // 16B aligned
      unsigned lds = lbase + (unsigned)((row * LPST + c4 * 4) * 4);  // 16B aligned
      asm volatile("global_load_async_to_lds_b128 %0, %1, off"
                   :: "v"(lds), "v"(g) : "memory");
    }
  };

  // prologue: chunk 0
  issue_fill(0, 0);
  asm volatile("s_wait_asynccnt 0" ::: "memory");
  __syncthreads();

  // t = 0: only l==0 alive with value log(probs[b,0,0]+eps)
  float v;
  {
    float raw = lpbuf[0][0];                   // broadcast read
    v = (tid == 0) ? flog(raw + 1e-30f) : NEGV;
  }
  if (ml == 0 && tid == tl) alpha_out[b] = v;
  sbuf[0][tid] = v;
  __syncthreads();
  int p = 0;                                   // sbuf[p] holds current state

  // t = 1: single peeled step so pairs start at even t
  {
    float lpv = flog(lpbuf[0][tid * LPST + 1] + 1e-30f);
    float vm1 = (tid > 0) ? sbuf[0][tid - 1] : NEGV;
    float nv = lae(v, vm1) + lpv;
    sbuf[1][tid] = nv;
    v = nv;
    if (ml == 1 && tid == tl) alpha_out[b] = nv;
    __syncthreads();
    p = 1;
  }

  const int rm1 = (tid > 0) ? (tid - 1) : 0;   // clamped neighbor row (value unused at tid==0)
  for (int c = 0; c < NCHUNK; ++c) {
    if (c + 1 < NCHUNK) issue_fill((c + 1) & 1, (c + 1) * TC);  // overlap DMA
    const float* lpc = lpbuf[c & 1];
#pragma unroll
    for (int tt = (c == 0) ? 2 : 0; tt < TC; tt += 2) {
      int t = c * TC + tt;                     // even; pair covers (t, t+1)
      // LDS reads: own lp pair (aligned b64) + neighbor lp scalar + 2 state values
      float2 lp2 = *(const float2*)&lpc[tid * LPST + tt];
      float lpn  = lpc[rm1 * LPST + tt];
      float am1  = (tid > 0) ? sbuf[p][tid - 1] : NEGV;
      float am2  = (tid > 1) ? sbuf[p][tid - 2] : NEGV;

      float lgA  = flog(lp2.x + 1e-30f);       // lp[l,   t]
      float lgB  = flog(lp2.y + 1e-30f);       // lp[l,   t+1]
      float lgA1 = flog(lpn  + 1e-30f);        // lp[l-1, t]

      float n1   = lae(v, am1) + lgA;                          // state[l]   at t
      float n1m1 = (tid > 0) ? (lae(am1, am2) + lgA1) : NEGV;  // state[l-1] at t
      float n2   = lae(n1, n1m1) + lgB;                        // state[l]   at t+1

      sbuf[p ^ 1][tid] = n2;
      v = n2;
      if (tid == tl) {
        if (ml == t)          alpha_out[b] = n1;
        else if (ml == t + 1) alpha_out[b] = n2;
      }
      if (tt == TC - 2)  // fold DMA fence into the last pair's barrier
        asm volatile("s_wait_asynccnt 0" ::: "memory");
      __syncthreads();
      p ^= 1;
    }
  }
}

// ---------------- Kernel 3: loss = -mean(alpha_last) (deterministic) ----------------
__global__ void mdn_reduce_kernel(const float* __restrict__ alpha,
                                  float* __restrict__ loss) {
  if (threadIdx.x == 0) {
    float s = 0.f;
    for (int b = 0; b < B_; ++b) s += alpha[b];
    loss[0] = -s * (1.0f / B_);
  }
}

extern "C" void kernel_launch(void* const* d_in, const int* in_sizes, int n_in,
                              void* d_out, int out_size, void* d_ws, size_t ws_size,
                              hipStream_t stream) {
  const float* probs    = (const float*)d_in[0];
  // d_in[1] = melspec: numerically unused by the reference
  const int*   text_len = (const int*)d_in[2];
  const int*   mel_len  = (const int*)d_in[3];

  float* out      = (float*)d_out;   // [0] = mdn_loss, [1..] = log_prob_matrix flat
  float* alpha_ws = (float*)d_ws;    // 64 floats

  const int n4 = (B_ * L_ * T_) / 4;                  // 6,553,600
  mdn_log_kernel<<<n4 / 256, 256, 0, stream>>>((const float4*)probs, out + 1, n4);
  mdn_dp_kernel<<<B_, L_, 0, stream>>>(probs, text_len, mel_len, alpha_ws);
  mdn_reduce_kernel<<<1, 32, 0, stream>>>(alpha_ws, out);
}